// SpectralConv1d_41257455846096
// MI455X (gfx1250) — compile-verified
//
#include <hip/hip_runtime.h>
#include <hip/hip_bf16.h>

#define B_     32
#define CIN_   64
#define COUT_  64
#define N_     8192
#define MODES_ 2048

typedef float v2f __attribute__((ext_vector_type(2)));
typedef float v8f __attribute__((ext_vector_type(8)));
typedef unsigned int u32x4 __attribute__((ext_vector_type(4)));
typedef int i32x4 __attribute__((ext_vector_type(4)));
typedef int i32x8 __attribute__((ext_vector_type(8)));

// ---------------------------------------------------------------------------
// Pass 1 / Pass 3: radix-2 transform replicating the reference's stage
// structure: blocks of 2*step; out[j] = e + cas[j]*o, out[j+step] = e - cas[j]*o,
// cas[j] = cos(2*pi*j/N) + sin(2*pi*j/N).
// One row per 256-thread block, staged in LDS. The global->LDS row copy is
// done by the Tensor Data Mover (one descriptor per block, issued by wave 0),
// overlapped with the in-LDS twiddle-table computation, synchronized via
// s_wait_tensorcnt + workgroup barrier.
// ---------------------------------------------------------------------------
__global__ __launch_bounds__(256) void fht_kernel(const float* __restrict__ src,
                                                  float* __restrict__ dst,
                                                  int in_len, int out_len,
                                                  float scale)
{
    __shared__ float X[N_];            // 32 KB row buffer
    __shared__ float casT[N_ / 2];     // 16 KB twiddle table (j < max step)
    const int tid = threadIdx.x;
    const int row = blockIdx.x;

    // --- TDM: async DMA of the row (in_len f32, contiguous) into LDS. ---
    // D# per CDNA5 ISA ch.8: 2D tile {in_len x 1}, data_size=4B, count=1,
    // type=2 ("image"). Issued once per block by wave 0 (TDM ignores EXEC).
    if (tid < 32) {
        const unsigned lds_off = (unsigned)(size_t)&X[0];
        const unsigned long long ga =
            (unsigned long long)(size_t)(src + (size_t)row * in_len);

        u32x4 g0;
        g0[0] = 1u;                                        // count=1, user D#
        g0[1] = lds_off;                                   // lds_addr [63:32]
        g0[2] = (unsigned)(ga & 0xffffffffu);              // global_addr lo
        g0[3] = (unsigned)((ga >> 32) & 0x01ffffffu)       // global_addr[56:32]
              | 0x80000000u;                               // type=2 [127:126]

        i32x8 g1;
        g1[0] = 2 << 16;                                   // data_size=4B, no mask
        g1[1] = (in_len & 0xffff) << 16;                   // tensor_dim0[15:0]
        g1[2] = ((in_len >> 16) & 0xffff) | (1 << 16);     // dim0 hi | tensor_dim1=1
        g1[3] = (in_len & 0xffff) << 16;                   // tile_dim0 = in_len
        g1[4] = 1;                                         // tile_dim1=1, tile_dim2=0
        g1[5] = in_len;                                    // tensor_dim0_stride lo
        g1[6] = 0;                                         // stride hi | dim1_stride lo
        g1[7] = 0;

        i32x4 g2;
        g2[0] = 1;                                         // tensor_dim2 = 1
        g2[1] = 1;                                         // tensor_dim3 = 1
        g2[2] = 0;
        g2[3] = 0;                                         // tile_dim3 = 0 (unused)
        i32x4 g3 = {0, 0, 0, 0};
        i32x8 g4 = {0, 0, 0, 0, 0, 0, 0, 0};               // extra group (zero)

        __builtin_amdgcn_tensor_load_to_lds(g0, g1, g2, g3, g4, 0);
    }

    // While the TDM runs: zero the padded tail (pass 3 only) ...
    {
        float4* X4 = (float4*)X;
        const int nin4 = in_len >> 2;
        for (int i4 = tid + nin4; i4 < N_ / 4; i4 += 256)
            X4[i4] = make_float4(0.f, 0.f, 0.f, 0.f);
    }
    // ... and fill the twiddle table: cas[j] for j in [0, N/2).
    {
        const float wconst = 6.28318530717958647692f / (float)N_;
        #pragma unroll
        for (int t = 0; t < N_ / 2 / 256; ++t) {
            int j = tid + t * 256;
            float a = wconst * (float)j;
            casT[j] = __cosf(a) + __sinf(a);
        }
    }
    if (tid < 32) __builtin_amdgcn_s_wait_tensorcnt(0);
    __syncthreads();

    // 13 butterfly stages; 4096 disjoint butterflies per stage, 16 per thread.
    int lg = 0;
    for (int step = 1; step < N_; step <<= 1, ++lg) {
        #pragma unroll
        for (int t = 0; t < N_ / 2 / 256; ++t) {
            int q  = tid + (t << 8);                 // butterfly id
            int j  = q & (step - 1);                 // offset within block
            int i0 = ((q >> lg) << (lg + 1)) + j;    // even element
            float c = casT[j];
            float e = X[i0];
            float o = X[i0 + step];
            X[i0]        = fmaf(c, o, e);
            X[i0 + step] = fmaf(-c, o, e);
        }
        __syncthreads();
    }

    // Vectorized store of the first out_len elements (2048 or 8192), scaled.
    {
        float4* d4 = (float4*)(dst + (size_t)row * out_len);
        const float4* X4 = (const float4*)X;
        const int nout4 = out_len >> 2;
        #pragma unroll
        for (int t = 0; t < N_ / 4 / 256; ++t) {
            int i4 = tid + t * 256;
            if (i4 < nout4) {
                float4 v = X4[i4];
                v.x *= scale; v.y *= scale; v.z *= scale; v.w *= scale;
                d4[i4] = v;
            }
        }
    }
}

// ---------------------------------------------------------------------------
// Pass 2: per-frequency GEMM  out[b,o,x] = sum_i A[b,i,x]*Wp[i,o,x]
//                                        + A[b,i,xn]*Wm[i,o,x]
// with Wp = 0.5*(W[x]+W[xn]), Wm = 0.5*(W[x]-W[xn]), xn = (M-x)%M.
// One wave32 per frequency; 8 f32 16x16 accumulator tiles (M=32 batch,
// N=64 cout), K=CIN=64 in steps of 4 via V_WMMA_F32_16X16X4_F32.
//
// Fragment layouts per CDNA5 ISA 7.12.2:
//   A (16x4):  lanes 0-15 -> M=lane, VGPR{0,1} = K{0,1}; lanes 16-31 -> K{2,3}
//   B (4x16):  lanes 0-15 -> N=lane, VGPR{0,1} = K{0,1}; lanes 16-31 -> K{2,3}
//   C/D (16x16): VGPR v -> M=v (lanes 0-15) / M=v+8 (lanes 16-31), N=lane%16
// ---------------------------------------------------------------------------
__global__ __launch_bounds__(256) void spectral_wmma(const float* __restrict__ xht,
                                                     const float* __restrict__ w,
                                                     float* __restrict__ om)
{
    const int lane = threadIdx.x & 31;
    const int wv   = threadIdx.x >> 5;
    const int x    = blockIdx.x * 8 + wv;          // frequency, 0..MODES-1
    const int xn   = (MODES_ - x) & (MODES_ - 1);  // negative frequency
    const int lh   = lane & 15;
    const int hi   = lane >> 4;                    // 0: K pair {0,1}, 1: {2,3}

    v8f acc[2][4] = {};

    for (int kk = 0; kk < CIN_; kk += 4) {
        const int k0 = kk + (hi << 1);             // this lane's first K index

        // A fragments (and negative-frequency copy) for both M tiles.
        v2f a[2], an[2];
        #pragma unroll
        for (int m = 0; m < 2; ++m) {
            const int brow = m * 16 + lh;
            const float* base = xht + ((size_t)brow * CIN_ + k0) * MODES_;
            a[m].x  = base[x];
            a[m].y  = base[MODES_ + x];            // K = k0+1
            an[m].x = base[xn];
            an[m].y = base[MODES_ + xn];
        }

        // B fragments: build Wp/Wm on the fly for all four N tiles.
        v2f bp[4], bm[4];
        #pragma unroll
        for (int n = 0; n < 4; ++n) {
            const int oc = n * 16 + lh;
            const float* w0p = w + ((size_t)k0 * COUT_ + oc) * MODES_;
            const float* w1p = w + ((size_t)(k0 + 1) * COUT_ + oc) * MODES_;
            float w0 = w0p[x], wn0 = w0p[xn];
            float w1 = w1p[x], wn1 = w1p[xn];
            bp[n].x = 0.5f * (w0 + wn0); bp[n].y = 0.5f * (w1 + wn1);
            bm[n].x = 0.5f * (w0 - wn0); bm[n].y = 0.5f * (w1 - wn1);
            // Prefetch next K-step's weight lines into cache (global_prefetch).
            if (kk + 4 < CIN_) {
                __builtin_prefetch(w0p + (size_t)4 * COUT_ * MODES_ + x, 0, 1);
                __builtin_prefetch(w0p + (size_t)4 * COUT_ * MODES_ + xn, 0, 1);
            }
        }

        #pragma unroll
        for (int m = 0; m < 2; ++m) {
            #pragma unroll
            for (int n = 0; n < 4; ++n) {
                acc[m][n] = __builtin_amdgcn_wmma_f32_16x16x4_f32(
                    false, a[m],  false, bp[n], (short)0, acc[m][n], false, false);
                acc[m][n] = __builtin_amdgcn_wmma_f32_16x16x4_f32(
                    false, an[m], false, bm[n], (short)0, acc[m][n], false, false);
            }
        }
    }

    // Scatter C/D tiles to om[b][o][x].
    #pragma unroll
    for (int m = 0; m < 2; ++m) {
        #pragma unroll
        for (int n = 0; n < 4; ++n) {
            #pragma unroll
            for (int v = 0; v < 8; ++v) {
                const int brow = m * 16 + v + hi * 8;
                const int oc   = n * 16 + lh;
                om[((size_t)brow * COUT_ + oc) * MODES_ + x] = acc[m][n][v];
            }
        }
    }
}

// ---------------------------------------------------------------------------
// Host-side launch: x -> xht (modes only) -> spectral WMMA GEMM -> inverse
// transform (zero-padded, scaled by 1/N).
// Workspace: xht = B*CIN*MODES f32 (16 MB), om = B*COUT*MODES f32 (16 MB).
// ---------------------------------------------------------------------------
extern "C" void kernel_launch(void* const* d_in, const int* in_sizes, int n_in,
                              void* d_out, int out_size, void* d_ws, size_t ws_size,
                              hipStream_t stream)
{
    const float* x = (const float*)d_in[0];   // (B, CIN, N)
    const float* w = (const float*)d_in[1];   // (CIN, COUT, MODES)
    float* out = (float*)d_out;               // (B, COUT, N)

    float* xht = (float*)d_ws;
    float* om  = xht + (size_t)B_ * CIN_ * MODES_;

    fht_kernel<<<B_ * CIN_, 256, 0, stream>>>(x, xht, N_, MODES_, 1.0f);
    spectral_wmma<<<MODES_ / 8, 256, 0, stream>>>(xht, w, om);
    fht_kernel<<<B_ * COUT_, 256, 0, stream>>>(om, out, MODES_, N_,
                                               1.0f / (float)N_);
}